// SuperpixelSAN_24446953849544
// MI455X (gfx1250) — compile-verified
//
#include <hip/hip_runtime.h>
#include <math.h>

typedef __attribute__((ext_vector_type(2)))  float v2f;
typedef __attribute__((ext_vector_type(8)))  float v8f;

#define NEG_SLOPE 0.2f

// ---------------------------------------------------------------------------
// Atomic helpers
// ---------------------------------------------------------------------------
__device__ __forceinline__ void atomAddF(float* p, float v) {
  // relaxed agent-scope fadd -> native global_atomic_add_f32
  __hip_atomic_fetch_add(p, v, __ATOMIC_RELAXED, __HIP_MEMORY_SCOPE_AGENT);
}

__device__ __forceinline__ void atomicMaxF(float* addr, float v) {
  // int/uint punning trick: valid for all finite floats with -inf init
  if (v >= 0.0f) atomicMax((int*)addr, __float_as_int(v));
  else           atomicMin((unsigned int*)addr, __float_as_uint(v));
}

// ---------------------------------------------------------------------------
// Utility kernels
// ---------------------------------------------------------------------------
__global__ __launch_bounds__(256) void fill_kernel(float* __restrict__ p, float v, long n) {
  long i = (long)blockIdx.x * blockDim.x + threadIdx.x;
  if (i < n) p[i] = v;
}

__global__ __launch_bounds__(256) void relu_kernel(float* __restrict__ p, long n) {
  long i = (long)blockIdx.x * blockDim.x + threadIdx.x;
  if (i < n) p[i] = fmaxf(p[i], 0.0f);
}

// ---------------------------------------------------------------------------
// Dense linear via CDNA5 WMMA:  out[r, oc_off+c] = sum_k x[r,k] * W[k,c] (+bias)
// FIN/FOUT/ACCUM compile-time. W + bias staged zero-padded in LDS.
// All addressing 32-bit. A loads clamped+selected (branch-free at wmma).
// One wave = 16 rows; 8 waves/block = 128 rows/block.
// ---------------------------------------------------------------------------
template <int FIN, int FOUT, bool ACCUM>
__global__ __launch_bounds__(256) void linear16_wmma(
    const float* __restrict__ x, int ldx,
    const float* __restrict__ W, const float* __restrict__ bias,
    float* __restrict__ out, int ldo, int oc_off, int n)
{
  constexpr int KT = (FIN + 3) / 4;    // K-tiles of 4
  constexpr int KP = KT * 4;           // padded K
  __shared__ float sW[KP * 16];        // [K][16] zero-padded
  __shared__ float sB[16];

  for (int t = threadIdx.x; t < KP * 16; t += 256) {
    const int k = t >> 4, cc = t & 15;
    sW[t] = (k < FIN && cc < FOUT) ? W[k * FOUT + cc] : 0.f;
  }
  if (threadIdx.x < 16)
    sB[threadIdx.x] = (!ACCUM && (int)threadIdx.x < FOUT) ? bias[threadIdx.x] : 0.f;
  __syncthreads();

  const int wave = threadIdx.x >> 5;
  const int lane = threadIdx.x & 31;
  const int half = lane >> 4;          // 0: lanes 0-15, 1: lanes 16-31
  const int l16  = lane & 15;
  const int rowBase = (blockIdx.x * 8 + wave) * 16;
  const int row = rowBase + l16;       // A-fragment row for this lane
  const bool rok = row < n;
  const int xbase = (rok ? row : n - 1) * ldx;   // always-valid 32-bit base

  v8f acc = {0.f, 0.f, 0.f, 0.f, 0.f, 0.f, 0.f, 0.f};

#pragma unroll
  for (int kt = 0; kt < KT; ++kt) {
    const int ka = kt * 4 + half * 2;            // lane-half K base
    // A 16x4 (ISA layout): VGPR0 = K {0|2}, VGPR1 = K {1|3}
    const int k0 = (ka + 0 < FIN) ? ka + 0 : FIN - 1;
    const int k1 = (ka + 1 < FIN) ? ka + 1 : FIN - 1;
    const float a0 = x[xbase + k0];              // unconditional loads
    const float a1 = x[xbase + k1];
    v2f a, b;
    a.x = (rok && (ka + 0) < FIN) ? a0 : 0.f;    // v_cndmask, no branch
    a.y = (rok && (ka + 1) < FIN) ? a1 : 0.f;
    // B 4x16 from zero-padded LDS: unconditional ds_load
    b.x = sW[(ka + 0) * 16 + l16];
    b.y = sW[(ka + 1) * 16 + l16];
    acc = __builtin_amdgcn_wmma_f32_16x16x4_f32(false, a, false, b,
                                                (short)0, acc, false, false);
  }

  // D 16x16 f32: VGPR i -> (M=i, N=l16) lanes 0-15 ; (M=i+8, N=l16) lanes 16-31
  if (l16 < FOUT) {
    const float bv = sB[l16];
    const int colOff = oc_off + l16;
    if (rowBase + 16 <= n) {                     // uniform fast path: full tile
#pragma unroll
      for (int i = 0; i < 8; ++i) {
        const int r = rowBase + i + half * 8;
        float* dst = &out[r * ldo + colOff];
        const float v = acc[i] + bv;
        if (ACCUM) *dst += v; else *dst = v;
      }
    } else {
#pragma unroll
      for (int i = 0; i < 8; ++i) {
        const int r = rowBase + i + half * 8;
        if (r < n) {
          float* dst = &out[r * ldo + colOff];
          const float v = acc[i] + bv;
          if (ACCUM) *dst += v; else *dst = v;
        }
      }
    }
  }
}

// ---------------------------------------------------------------------------
// Per-node GAT attention scalars: hd[i] = h[i,:].ag[0:15], hs[i] = h[i,:].ag[15:30]
// h stored with leading dim 16 (64B-aligned rows -> b128 loads).
// ---------------------------------------------------------------------------
__global__ __launch_bounds__(256) void node_scalars(
    const float* __restrict__ h, const float* __restrict__ ag,
    float* __restrict__ hdv, float* __restrict__ hsv, int n)
{
  int i = blockIdx.x * blockDim.x + threadIdx.x;
  if (i >= n) return;
  const float4* h4 = (const float4*)(h + (size_t)i * 16);
  float hv[16];
  ((float4*)hv)[0] = h4[0]; ((float4*)hv)[1] = h4[1];
  ((float4*)hv)[2] = h4[2]; ((float4*)hv)[3] = h4[3];
  float a = 0.f, b = 0.f;
#pragma unroll
  for (int f = 0; f < 15; ++f) { a += hv[f] * ag[f]; b += hv[f] * ag[15 + f]; }
  hdv[i] = a; hsv[i] = b;
}

// ---------------------------------------------------------------------------
// Edge pass 1: e = leaky_relu(hd[dst] + hs[src]); segment-max into m[dst]
// ---------------------------------------------------------------------------
__global__ __launch_bounds__(256) void edge_pass1(
    const int* __restrict__ idx, const float* __restrict__ hdv,
    const float* __restrict__ hsv, float* __restrict__ ebuf,
    float* __restrict__ m, int e)
{
  int i = blockIdx.x * blockDim.x + threadIdx.x;
  if (i >= e) return;
  const int d = idx[i], s = idx[e + i];
  float v = hdv[d] + hsv[s];
  v = (v > 0.f) ? v : NEG_SLOPE * v;
  ebuf[i] = v;
  atomicMaxF(&m[d], v);
}

// ---------------------------------------------------------------------------
// Edge pass 2: w = exp(e - m[dst]); segment-sum into s[dst]; store w in-place
// ---------------------------------------------------------------------------
__global__ __launch_bounds__(256) void edge_pass2(
    const int* __restrict__ idx, float* __restrict__ ebuf,
    const float* __restrict__ m, float* __restrict__ sden, int e)
{
  int i = blockIdx.x * blockDim.x + threadIdx.x;
  if (i >= e) return;
  const int d = idx[i];
  const float w = expf(ebuf[i] - m[d]);
  ebuf[i] = w;
  atomAddF(&sden[d], w);
}

// ---------------------------------------------------------------------------
// Edge pass 3: fused GAT aggregate + SpMM:
//   out[dst, off+f] += (w/s[dst]) * h[src,f] + val * p[src,f]
// h/p rows gathered as 4x global_load_b128 each (64B-aligned rows).
// ---------------------------------------------------------------------------
__global__ __launch_bounds__(256) void edge_pass3(
    const int* __restrict__ idx, const float* __restrict__ val,
    const float* __restrict__ ebuf, const float* __restrict__ sden,
    const float* __restrict__ h, const float* __restrict__ p,
    float* __restrict__ xn, int off, int e)
{
  int i = blockIdx.x * blockDim.x + threadIdx.x;
  if (i >= e) return;
  const int d = idx[i], s = idx[e + i];
  const float alpha = ebuf[i] / fmaxf(sden[d], 1e-16f);
  const float vv = val[i];
  const float4* h4 = (const float4*)(h + (size_t)s * 16);
  const float4* p4 = (const float4*)(p + (size_t)s * 16);
  float hv[16], pv[16];
  ((float4*)hv)[0] = h4[0]; ((float4*)hv)[1] = h4[1];
  ((float4*)hv)[2] = h4[2]; ((float4*)hv)[3] = h4[3];
  ((float4*)pv)[0] = p4[0]; ((float4*)pv)[1] = p4[1];
  ((float4*)pv)[2] = p4[2]; ((float4*)pv)[3] = p4[3];
  float* dst = xn + (size_t)d * 30 + off;
#pragma unroll
  for (int f = 0; f < 15; ++f)
    atomAddF(&dst[f], alpha * hv[f] + vv * pv[f]);
}

// ---------------------------------------------------------------------------
// Mean-pool accumulation per (level, graph)
// ---------------------------------------------------------------------------
__global__ __launch_bounds__(256) void pool_kernel(
    const float* __restrict__ x4, const int* __restrict__ batch,
    float* __restrict__ pool, float* __restrict__ cnt, int n, int L)
{
  int i = blockIdx.x * blockDim.x + threadIdx.x;
  if (i >= n) return;
  const int b = batch[i];
  float* pr = pool + ((long)L * 64 + b) * 10;
  const float* xr = x4 + (long)i * 10;
#pragma unroll
  for (int c = 0; c < 10; ++c) atomAddF(&pr[c], xr[c]);
  atomAddF(&cnt[L * 64 + b], 1.0f);
}

// ---------------------------------------------------------------------------
// Final: divide pools, combine 30->10, softmax. One thread per graph (B=64).
// ---------------------------------------------------------------------------
__global__ __launch_bounds__(64) void finalize_kernel(
    const float* __restrict__ pool, const float* __restrict__ cnt,
    const float* __restrict__ combW, const float* __restrict__ combb,
    float* __restrict__ out)
{
  int b = blockIdx.x * blockDim.x + threadIdx.x;
  if (b >= 64) return;
  float xc[30];
#pragma unroll
  for (int L = 0; L < 3; ++L) {
    const float c = fmaxf(cnt[L * 64 + b], 1.0f);
    for (int j = 0; j < 10; ++j) xc[L * 10 + j] = pool[((long)L * 64 + b) * 10 + j] / c;
  }
  float lg[10]; float mx = -INFINITY;
#pragma unroll
  for (int j = 0; j < 10; ++j) {
    float s = combb[j];
    for (int k = 0; k < 30; ++k) s += xc[k] * combW[k * 10 + j];
    lg[j] = s; mx = fmaxf(mx, s);
  }
  float den = 0.f;
#pragma unroll
  for (int j = 0; j < 10; ++j) { lg[j] = expf(lg[j] - mx); den += lg[j]; }
#pragma unroll
  for (int j = 0; j < 10; ++j) out[b * 10 + j] = lg[j] / den;
}

// ---------------------------------------------------------------------------
// Host orchestration
// ---------------------------------------------------------------------------
static inline unsigned cdivl(long a, long b) { return (unsigned)((a + b - 1) / b); }

extern "C" void kernel_launch(void* const* d_in, const int* in_sizes, int n_in,
                              void* d_out, int out_size, void* d_ws, size_t ws_size,
                              hipStream_t stream) {
  (void)n_in; (void)out_size; (void)ws_size;

  // ---- input map (top-level insertion order; params pytree sorted-key) ----
  const float* X[3]     = { (const float*)d_in[0], (const float*)d_in[1], (const float*)d_in[2] };
  const int*   batch[3] = { (const int*)d_in[11], (const int*)d_in[12], (const int*)d_in[13] };
  int nlev[3] = { in_sizes[0] / 3, in_sizes[1] / 3, in_sizes[2] / 3 };

  const int* eidx[3][2]; const float* eval_[3][2]; int ecnt[3];
  eidx[0][0] = eidx[0][1] = (const int*)d_in[3];
  eval_[0][0] = eval_[0][1] = (const float*)d_in[4];  ecnt[0] = in_sizes[4];
  eidx[1][0] = (const int*)d_in[5];  eval_[1][0] = (const float*)d_in[6];
  eidx[1][1] = (const int*)d_in[7];  eval_[1][1] = (const float*)d_in[8];  ecnt[1] = in_sizes[6];
  eidx[2][0] = eidx[2][1] = (const int*)d_in[9];
  eval_[2][0] = eval_[2][1] = (const float*)d_in[10]; ecnt[2] = in_sizes[10];

  const float* combW = (const float*)d_in[14];
  const float* combb = (const float*)d_in[15];

  long nmax = nlev[0]; if (nlev[1] > nmax) nmax = nlev[1]; if (nlev[2] > nmax) nmax = nlev[2];
  long emax = ecnt[0]; if (ecnt[1] > emax) emax = ecnt[1]; if (ecnt[2] > emax) emax = ecnt[2];

  // ---- carve workspace ----
  char* w = (char*)d_ws;
  auto carve = [&](size_t nfloats) -> float* {
    float* p = (float*)w;
    w += ((nfloats * sizeof(float)) + 255) & ~(size_t)255;
    return p;
  };
  float* x1    = carve(nmax * 30);
  float* x2    = carve(nmax * 30);
  float* x3    = carve(nmax * 30);
  float* hbuf  = carve(nmax * 16);
  float* pbuf  = carve(nmax * 16);
  float* hdv   = carve(nmax);
  float* hsv   = carve(nmax);
  float* mbuf  = carve(nmax);
  float* sbuf  = carve(nmax);
  float* ebuf  = carve(emax);
  float* x4    = carve(nmax * 10);
  float* pool  = carve(3 * 64 * 10);
  float* cntb  = carve(3 * 64);

  const int T = 256;
  fill_kernel<<<cdivl(3 * 64 * 10, T), T, 0, stream>>>(pool, 0.f, 3 * 64 * 10);
  fill_kernel<<<cdivl(3 * 64, T), T, 0, stream>>>(cntb, 0.f, 3 * 64);

  for (int L = 0; L < 3; ++L) {
    const int pb = 16 + L * 32;
    const float* finW = (const float*)d_in[pb + 0];
    const float* finb = (const float*)d_in[pb + 1];
    const int nn = nlev[L];
    const int ee = ecnt[L];
    const unsigned gN  = cdivl(nn, T);
    const unsigned gE  = cdivl(ee, T);
    const unsigned gW  = cdivl(nn, 128);   // 128 rows / block in WMMA linear

    const float* xin = X[L]; int ldx = 3;
    float* xl[3] = { x1, x2, x3 };

    for (int layer = 0; layer < 3; ++layer) {
      float* xn = xl[layer];
      fill_kernel<<<cdivl((long)nn * 30, T), T, 0, stream>>>(xn, 0.f, (long)nn * 30);

      for (int k = 0; k < 2; ++k) {
        const int hp = pb + 2 + (layer * 2 + k) * 5;
        const float* Wg = (const float*)d_in[hp + 0];
        const float* Wp = (const float*)d_in[hp + 1];
        const float* ag = (const float*)d_in[hp + 2];
        const float* bg = (const float*)d_in[hp + 3];
        const float* bp = (const float*)d_in[hp + 4];

        if (layer == 0) {
          linear16_wmma<3, 15, false><<<gW, T, 0, stream>>>(xin, ldx, Wg, bg, hbuf, 16, 0, nn);
          linear16_wmma<3, 15, false><<<gW, T, 0, stream>>>(xin, ldx, Wp, bp, pbuf, 16, 0, nn);
        } else {
          linear16_wmma<30, 15, false><<<gW, T, 0, stream>>>(xin, ldx, Wg, bg, hbuf, 16, 0, nn);
          linear16_wmma<30, 15, false><<<gW, T, 0, stream>>>(xin, ldx, Wp, bp, pbuf, 16, 0, nn);
        }
        node_scalars<<<gN, T, 0, stream>>>(hbuf, ag, hdv, hsv, nn);
        fill_kernel<<<gN, T, 0, stream>>>(mbuf, -INFINITY, nn);
        fill_kernel<<<gN, T, 0, stream>>>(sbuf, 0.f, nn);
        edge_pass1<<<gE, T, 0, stream>>>(eidx[L][k], hdv, hsv, ebuf, mbuf, ee);
        edge_pass2<<<gE, T, 0, stream>>>(eidx[L][k], ebuf, mbuf, sbuf, ee);
        edge_pass3<<<gE, T, 0, stream>>>(eidx[L][k], eval_[L][k], ebuf, sbuf,
                                         hbuf, pbuf, xn, k * 15, ee);
      }
      relu_kernel<<<cdivl((long)nn * 30, T), T, 0, stream>>>(xn, (long)nn * 30);
      xin = xn; ldx = 30;
    }

    // x4 = [x1|x2|x3] @ finW(90x10) + finb  ==  x1@W[0:30] + x2@W[30:60] + x3@W[60:90]
    linear16_wmma<30, 10, false><<<gW, T, 0, stream>>>(x1, 30, finW,       finb,    x4, 10, 0, nn);
    linear16_wmma<30, 10, true ><<<gW, T, 0, stream>>>(x2, 30, finW + 300, nullptr, x4, 10, 0, nn);
    linear16_wmma<30, 10, true ><<<gW, T, 0, stream>>>(x3, 30, finW + 600, nullptr, x4, 10, 0, nn);
    pool_kernel<<<gN, T, 0, stream>>>(x4, batch[L], pool, cntb, nn, L);
  }

  finalize_kernel<<<1, 64, 0, stream>>>(pool, cntb, combW, combb, (float*)d_out);
}